// customConv_89498528514358
// MI455X (gfx1250) — compile-verified
//
#include <hip/hip_runtime.h>

// ---------------------------------------------------------------------------
// 5x5 masked conv == dilated 3x3 (dilation-2) GEMM + constant neighbor stencil
// Target: gfx1250 (CDNA5), wave32, V_WMMA_F32_16X16X4_F32 (exact fp32 path).
//
// - PX=64: each wave computes 16 oc x 64 px (4 accumulators); each A operand
//   feeds 4 WMMAs.
// - Input tile stored channel-pair interleaved in LDS so each WMMA K-pair is
//   a single conflict-free ds_load_b64 (slab stride padded to 32-bank offset).
// - Packed weights stored as [tap][ic/2][oc][2] so each A K-pair is a single
//   coalesced global_load_b64.
// - Neighbor term: channel-summed tile T[5][68] first (b64 pair reads), then
//   16-tap stencil on T -> S[p]; fused into the store epilogue.
// ---------------------------------------------------------------------------

typedef float v2f __attribute__((ext_vector_type(2)));
typedef float v8f __attribute__((ext_vector_type(8)));

#define NB   8
#define CIN  64
#define HH   128
#define WW   128
#define COUT 128
#define PX   64                  // pixels per block (four 16-wide N-tiles)
#define LDW  (PX + 4)            // tile width with halo = 68
#define KTOT (9 * CIN)           // GEMM K: 9 dilated taps x 64 channels
#define NCP  (CIN / 2)           // channel pairs = 32
#define SLAB (5 * LDW * 2 + 56)  // floats per channel-pair slab; 736 mod 64
                                 // = 32 -> adjacent slabs hit disjoint banks

static constexpr float MUW = (1.0f - 0.9f) / 3.0f;

// wpack[(((tap*NCP + icp)*COUT) + oc)*2 + par] = weight[oc][icp*2+par][kh][kw]
// with (kh,kw) the tap's (even,even) position.
__global__ __launch_bounds__(256)
void pack_weights_kernel(const float* __restrict__ weight,
                         float* __restrict__ wpack) {
    int idx = blockIdx.x * 256 + threadIdx.x;        // 0 .. KTOT*COUT-1
    if (idx >= KTOT * COUT) return;
    int par = idx & 1;
    int oc  = (idx >> 1) & (COUT - 1);
    int kk  = idx >> 8;           // tap*NCP + icp
    int icp = kk & (NCP - 1);
    int tap = kk >> 5;            // 0..8
    int ic  = icp * 2 + par;
    int kh = (tap / 3) * 2;       // 0,2,4
    int kw = (tap % 3) * 2;       // 0,2,4
    wpack[idx] = weight[((oc * CIN + ic) * 5 + kh) * 5 + kw];
}

template <bool PACKED>
__global__ __launch_bounds__(256)
void conv_wmma_kernel(const float* __restrict__ input_,
                      const float* __restrict__ weight,
                      const float* __restrict__ wpack,
                      float* __restrict__ out) {
    __shared__ float lin[NCP * SLAB];      // pair-interleaved tile, ~94 KB
    __shared__ float sT[5 * LDW];          // channel-summed tile rows
    __shared__ float sS[PX];               // mu' * neighbor sum per pixel

    const int blk = blockIdx.x;            // N*H*(W/PX) = 2048 blocks
    const int w0  = (blk & 1) * PX;
    const int h   = (blk >> 1) & (HH - 1);
    const int n   = blk >> 8;
    const int t   = threadIdx.x;

    // ---- Stage input tile (rows h-2..h+2, cols w0-2..w0+PX+1), zero-padded,
    //      into pair-interleaved layout: (ic,r,c) ->
    //      (ic>>1)*SLAB + ((r*LDW + c)<<1) + (ic&1).
    const int TILE = CIN * 5 * LDW;        // 21760 elements
    for (int e = t; e < TILE; e += 256) {
        int ic  = e / (5 * LDW);
        int rem = e - ic * (5 * LDW);
        int r   = rem / LDW;
        int c   = rem - r * LDW;
        int gy = h + r - 2;
        int gx = w0 + c - 2;
        float v = 0.0f;
        if ((unsigned)gy < (unsigned)HH && (unsigned)gx < (unsigned)WW)
            v = input_[((n * CIN + ic) * HH + gy) * WW + gx];
        lin[(ic >> 1) * SLAB + ((r * LDW + c) << 1) + (ic & 1)] = v;
    }
    __syncthreads();

    // ---- Channel-summed tile: T[cell] = sum_ic tile[ic][cell], cell=r*LDW+c.
    for (int cell = t; cell < 5 * LDW; cell += 256) {
        float s = 0.0f;
        #pragma unroll 8
        for (int icp = 0; icp < NCP; ++icp) {
            v2f pr = *(const v2f*)&lin[icp * SLAB + (cell << 1)];
            s += pr.x + pr.y;
        }
        sT[cell] = s;
    }
    __syncthreads();

    // ---- Neighbor term from T: rows 1,3 all 5 cols; rows 0,2,4 cols p+1,p+3.
    if (t < PX) {
        const int p = t;
        float s = 0.0f;
        #pragma unroll
        for (int kw = 0; kw < 5; ++kw)
            s += sT[1 * LDW + p + kw] + sT[3 * LDW + p + kw];
        s += sT[0 * LDW + p + 1] + sT[0 * LDW + p + 3];
        s += sT[2 * LDW + p + 1] + sT[2 * LDW + p + 3];
        s += sT[4 * LDW + p + 1] + sT[4 * LDW + p + 3];
        sS[p] = MUW * s;
    }
    __syncthreads();

    // ---- GEMM: D[16oc x 16px] += A[16oc x 4k] * B[4k x 16px], fp32 WMMA.
    const int lane = t & 31;
    const int wv   = t >> 5;         // 8 waves
    const int ocb  = wv * 16;        // this wave's 16 output channels
    const int mrow = lane & 15;      // A: M row / B,C: N col
    const int hi   = lane >> 4;      // K-pair select (ISA A/B VGPR layout)

    v8f acc[4];                      // px tiles: w0 + 16*q + (0..15)
    #pragma unroll
    for (int q = 0; q < 4; ++q) acc[q] = (v8f){};

    for (int tap = 0; tap < 9; ++tap) {
        const int kh = (tap / 3) * 2;
        const int kw = (tap % 3) * 2;
        const int bcol = (kh * LDW + kw + mrow) << 1;   // pair offset in slab
        #pragma unroll
        for (int ic4 = 0; ic4 < 16; ++ic4) {
            const int icp = ic4 * 2 + hi;    // lane's channel pair
            v2f a;
            if (PACKED) {
                a = *(const v2f*)
                    &wpack[(((tap * NCP + icp) * COUT) + ocb + mrow) * 2];
            } else {
                const float* ap =
                    &weight[((ocb + mrow) * CIN + icp * 2) * 25 + kh * 5 + kw];
                a.x = ap[0];
                a.y = ap[25];
            }
            #pragma unroll
            for (int q = 0; q < 4; ++q) {
                v2f b = *(const v2f*)&lin[icp * SLAB + bcol + (q << 5)];
                acc[q] = __builtin_amdgcn_wmma_f32_16x16x4_f32(
                             false, a, false, b, (short)0, acc[q],
                             false, false);
            }
        }
    }

    // ---- Epilogue: D + mu' * S  ->  out[n][oc][h][w]
    float add[4];
    #pragma unroll
    for (int q = 0; q < 4; ++q) add[q] = sS[mrow + q * 16];
    #pragma unroll
    for (int j = 0; j < 8; ++j) {
        const int oc = ocb + j + 8 * hi;         // C/D layout: M = j + 8*hi
        float* o = &out[((n * COUT + oc) * HH + h) * WW + w0];
        #pragma unroll
        for (int q = 0; q < 4; ++q)
            o[mrow + q * 16] = acc[q][j] + add[q];
    }
}

extern "C" void kernel_launch(void* const* d_in, const int* in_sizes, int n_in,
                              void* d_out, int out_size, void* d_ws,
                              size_t ws_size, hipStream_t stream) {
    const float* input_ = (const float*)d_in[0];
    const float* weight = (const float*)d_in[1];
    float* out = (float*)d_out;

    const int nblocks = NB * HH * (WW / PX);     // 2048
    const size_t need = (size_t)KTOT * COUT * sizeof(float);  // 288 KB

    if (d_ws != nullptr && ws_size >= need) {
        float* wpack = (float*)d_ws;
        pack_weights_kernel<<<(KTOT * COUT + 255) / 256, 256, 0, stream>>>(
            weight, wpack);
        conv_wmma_kernel<true><<<nblocks, 256, 0, stream>>>(
            input_, weight, wpack, out);
    } else {
        conv_wmma_kernel<false><<<nblocks, 256, 0, stream>>>(
            input_, weight, nullptr, out);
    }
}